// GNNModel_55714315763894
// MI455X (gfx1250) — compile-verified
//
#include <hip/hip_runtime.h>
#include <math.h>

#define N_NODES 100000
#define N_EDGES 1600000
#define D_IN 128
#define D_H 64
#define D_OUT 32
#define BN_EPS 1e-5f

typedef __attribute__((ext_vector_type(2))) float v2f;
typedef __attribute__((ext_vector_type(8))) float v8f;

// ---------------------------------------------------------------------------
// Degree / normalization
// ---------------------------------------------------------------------------
__global__ void k_deg_init(float* deg, int n) {
    int i = blockIdx.x * blockDim.x + threadIdx.x;
    if (i < n) deg[i] = 1.0f;   // self-loop
}

__global__ void k_deg_edge(const int* __restrict__ dst, float* deg, int e) {
    int i = blockIdx.x * blockDim.x + threadIdx.x;
    if (i < e) unsafeAtomicAdd(&deg[dst[i]], 1.0f);
}

__global__ void k_dis(float* deg_dis, int n) {
    int i = blockIdx.x * blockDim.x + threadIdx.x;
    if (i < n) deg_dis[i] = rsqrtf(deg_dis[i]);  // in place: deg -> deg^-1/2
}

__global__ void k_norm(const int* __restrict__ src, const int* __restrict__ dst,
                       const float* __restrict__ dis, float* __restrict__ norm, int e) {
    int i = blockIdx.x * blockDim.x + threadIdx.x;
    if (i < e) norm[i] = dis[src[i]] * dis[dst[i]];
}

// ---------------------------------------------------------------------------
// Dense GEMM via V_WMMA_F32_16X16X4_F32.
// One wave computes a 16-row x NOUT-col strip of H = X(nrows x K) * W(K x NOUT).
// W is staged in LDS *K-pair-packed*: lw[(k/2)*2*NOUT + 2*c + (k&1)], so each
// WMMA B operand (W[kk][c], W[kk+1][c]) is a single aligned ds_load_b64.
// Block = 256 threads = 8 waves = 8 row strips.
// A 16x4 f32 layout: lanes 0-15 hold rows, VGPR0=K0 (half0)/K2 (half1), VGPR1=K1/K3.
// B 4x16 f32 layout: lanes hold cols, VGPR0=K0/K2, VGPR1=K1/K3.
// C/D 16x16 f32: VGPR v -> row v (lanes 0-15) / row v+8 (lanes 16-31).
// ---------------------------------------------------------------------------
template <int K, int NOUT>
__global__ void k_gemm_wmma(const float* __restrict__ X, const float* __restrict__ W,
                            float* __restrict__ H, int nrows) {
    constexpr int NT = NOUT / 16;
    __shared__ float lw[K * NOUT];

    // cooperative load of W into LDS, packing K-adjacent pairs per column
    for (int i = threadIdx.x; i < K * NOUT; i += blockDim.x) {
        int k = i / NOUT;            // NOUT is a power of two -> shift
        int c = i - k * NOUT;
        lw[(k >> 1) * (2 * NOUT) + 2 * c + (k & 1)] = W[i];
    }
    __syncthreads();

    const int wave = threadIdx.x >> 5;
    const int lane = threadIdx.x & 31;
    const int half = lane >> 4;      // 0: lanes 0-15, 1: lanes 16-31
    const int l    = lane & 15;

    const int strip = blockIdx.x * 8 + wave;
    const int rows  = strip * 16;
    if (rows >= nrows) return;       // uniform per wave -> EXEC stays all-ones

    const int row = rows + l;
    const v2f* lwp = (const v2f*)lw; // pair-row stride = NOUT pairs

    v8f acc[NT] = {};

#pragma unroll
    for (int k = 0; k < K; k += 4) {
        const int kk = k + 2 * half;
        v2f a;
        a.x = X[row * K + kk];
        a.y = X[row * K + kk + 1];
        const int prow = (kk >> 1) * NOUT;
#pragma unroll
        for (int t = 0; t < NT; ++t) {
            v2f b = lwp[prow + t * 16 + l];   // single ds_load_b64
            acc[t] = __builtin_amdgcn_wmma_f32_16x16x4_f32(
                false, a, false, b, (short)0, acc[t], false, false);
        }
    }

#pragma unroll
    for (int t = 0; t < NT; ++t) {
#pragma unroll
        for (int v = 0; v < 8; ++v) {
            H[(rows + v + 8 * half) * NOUT + t * 16 + l] = acc[t][v];
        }
    }
}

// ---------------------------------------------------------------------------
// Aggregation: agg[n][c] = bias[c] + h[n][c]*dis[n]^2  (self-loop term)
// then atomic scatter over edges: agg[dst] += h[src] * norm[e]
// ---------------------------------------------------------------------------
__global__ void k_msg_init(const float* __restrict__ h, const float* __restrict__ dis,
                           const float* __restrict__ bias, float* __restrict__ agg,
                           int n, int d) {
    int i = blockIdx.x * blockDim.x + threadIdx.x;
    if (i >= n * d) return;
    int node = i / d;
    int c = i - node * d;
    float s = dis[node];
    agg[i] = bias[c] + h[i] * (s * s);
}

// each thread: one edge, 4 channels (float4 gather, 4 fp32 atomics)
__global__ void k_msg_edge(const float* __restrict__ h, const float* __restrict__ norm,
                           const int* __restrict__ src, const int* __restrict__ dst,
                           float* __restrict__ agg, int e, int d, int qshift) {
    int gid = blockIdx.x * blockDim.x + threadIdx.x;
    int edge = gid >> qshift;
    if (edge >= e) return;
    int q4 = (gid & ((1 << qshift) - 1)) << 2;   // channel offset
    int s = src[edge];
    int t = dst[edge];
    float w = norm[edge];
    float4 hv = *(const float4*)(h + s * d + q4);
    float* base = agg + t * d + q4;
    unsafeAtomicAdd(base + 0, hv.x * w);
    unsafeAtomicAdd(base + 1, hv.y * w);
    unsafeAtomicAdd(base + 2, hv.z * w);
    unsafeAtomicAdd(base + 3, hv.w * w);
}

// ---------------------------------------------------------------------------
// BatchNorm (batch statistics) + ReLU
// ---------------------------------------------------------------------------
__global__ void k_zero_stats(float* stats) {
    if (threadIdx.x < 128) stats[threadIdx.x] = 0.0f;   // sum[64], sumsq[64]
}

__global__ void k_bn_stats(const float* __restrict__ agg, float* gsum, float* gsq, int n) {
    __shared__ float ls[256];
    __shared__ float lq[256];
    const int c = threadIdx.x & 63;
    const int j = threadIdx.x >> 6;     // 0..3 row groups per block
    float s = 0.0f, q = 0.0f;
    for (int row = blockIdx.x * 4 + j; row < n; row += gridDim.x * 4) {
        float v = agg[row * 64 + c];
        s += v;
        q += v * v;
    }
    ls[threadIdx.x] = s;
    lq[threadIdx.x] = q;
    __syncthreads();
    if (threadIdx.x < 64) {
        s = ls[c] + ls[c + 64] + ls[c + 128] + ls[c + 192];
        q = lq[c] + lq[c + 64] + lq[c + 128] + lq[c + 192];
        unsafeAtomicAdd(&gsum[c], s);
        unsafeAtomicAdd(&gsq[c], q);
    }
}

__global__ void k_bn_final(const float* gsum, const float* gsq,
                           const float* __restrict__ gamma, const float* __restrict__ beta,
                           float* scale, float* shift, int n) {
    int c = threadIdx.x;
    if (c >= 64) return;
    float inv_n = 1.0f / (float)n;
    float mean = gsum[c] * inv_n;
    float var  = gsq[c] * inv_n - mean * mean;
    float inv  = rsqrtf(var + BN_EPS);
    float sc   = gamma[c] * inv;
    scale[c] = sc;
    shift[c] = beta[c] - mean * sc;
}

__global__ void k_bn_apply(const float* __restrict__ agg, const float* __restrict__ scale,
                           const float* __restrict__ shift, float* __restrict__ out, int total) {
    int i = blockIdx.x * blockDim.x + threadIdx.x;
    if (i >= total) return;
    int c = i & 63;
    float v = agg[i] * scale[c] + shift[c];
    out[i] = fmaxf(v, 0.0f);
}

// ---------------------------------------------------------------------------
// Launcher
// ---------------------------------------------------------------------------
extern "C" void kernel_launch(void* const* d_in, const int* in_sizes, int n_in,
                              void* d_out, int out_size, void* d_ws, size_t ws_size,
                              hipStream_t stream) {
    const float* x      = (const float*)d_in[0];
    const int*   ei     = (const int*)d_in[1];
    const float* W1     = (const float*)d_in[2];
    const float* b1     = (const float*)d_in[3];
    const float* gamma1 = (const float*)d_in[4];
    const float* beta1  = (const float*)d_in[5];
    const float* W2     = (const float*)d_in[6];
    const float* b2     = (const float*)d_in[7];
    float* out = (float*)d_out;

    const int N = N_NODES, E = N_EDGES;
    const int* src = ei;
    const int* dst = ei + E;

    // workspace carve-up (256B aligned)
    char*  ws  = (char*)d_ws;
    size_t off = 0;
    auto carve = [&](size_t bytes) -> float* {
        off = (off + 255) & ~(size_t)255;
        float* p = (float*)(ws + off);
        off += bytes;
        return p;
    };
    float* dis   = carve((size_t)N * 4);         // deg -> deg^-1/2 (in place)
    float* norm  = carve((size_t)E * 4);
    float* h1    = carve((size_t)N * D_H * 4);   // reused as hrelu
    float* agg1  = carve((size_t)N * D_H * 4);   // reused as h2
    float* stats = carve(256 * 4);               // sum,sq,scale,shift
    float* gsum  = stats;
    float* gsq   = stats + 64;
    float* scale = stats + 128;
    float* shift = stats + 192;
    float* hrelu = h1;
    float* h2    = agg1;

    const int T = 256;

    // normalization coefficients
    k_deg_init<<<(N + T - 1) / T, T, 0, stream>>>(dis, N);
    k_deg_edge<<<(E + T - 1) / T, T, 0, stream>>>(dst, dis, E);
    k_dis<<<(N + T - 1) / T, T, 0, stream>>>(dis, N);
    k_norm<<<(E + T - 1) / T, T, 0, stream>>>(src, dst, dis, norm, E);

    // layer 1: GEMM (WMMA) -> aggregate -> BN -> ReLU
    const int strips = (N + 15) / 16;                 // 6250
    const int gblk   = (strips + 7) / 8;              // 8 waves / block
    k_gemm_wmma<D_IN, D_H><<<gblk, T, 0, stream>>>(x, W1, h1, N);

    k_msg_init<<<(N * D_H + T - 1) / T, T, 0, stream>>>(h1, dis, b1, agg1, N, D_H);
    k_msg_edge<<<(E * (D_H / 4) + T - 1) / T, T, 0, stream>>>(h1, norm, src, dst, agg1,
                                                              E, D_H, 4);

    k_zero_stats<<<1, 128, 0, stream>>>(stats);
    k_bn_stats<<<512, T, 0, stream>>>(agg1, gsum, gsq, N);
    k_bn_final<<<1, 64, 0, stream>>>(gsum, gsq, gamma1, beta1, scale, shift, N);
    k_bn_apply<<<(N * D_H + T - 1) / T, T, 0, stream>>>(agg1, scale, shift, hrelu, N * D_H);

    // layer 2: GEMM (WMMA) -> aggregate into d_out
    k_gemm_wmma<D_H, D_OUT><<<gblk, T, 0, stream>>>(hrelu, W2, h2, N);

    k_msg_init<<<(N * D_OUT + T - 1) / T, T, 0, stream>>>(h2, dis, b2, out, N, D_OUT);
    k_msg_edge<<<(E * (D_OUT / 4) + T - 1) / T, T, 0, stream>>>(h2, norm, src, dst, out,
                                                                E, D_OUT, 3);
}